// HypergraphEncoder_35467839931095
// MI455X (gfx1250) — compile-verified
//
#include <hip/hip_runtime.h>
#include <hip/hip_bf16.h>

#define N_NODES   100000
#define N_HEDGES  25000
#define N_EDGES   1000000
#define NUM_GRAPHS 64
#define H 64

typedef __attribute__((ext_vector_type(2))) float v2f;
typedef __attribute__((ext_vector_type(8))) float v8f;

// ---------------- utility ----------------
__global__ void fill_zero_kernel(float* p, int n) {
    int i = blockIdx.x * blockDim.x + threadIdx.x;
    if (i < n) p[i] = 0.0f;
}

// ---------------- encoders ----------------
// xn = relu(node_x @ Wn + bn); node_x is [N,1]
__global__ void encode_nodes_kernel(const float* __restrict__ nx,
                                    const float* __restrict__ Wn,
                                    const float* __restrict__ bn,
                                    float* __restrict__ xn, int n) {
    int i = blockIdx.x * blockDim.x + threadIdx.x;
    if (i >= n * H) return;
    int r = i >> 6, h = i & 63;
    float v = nx[r] * Wn[h] + bn[h];
    xn[i] = fmaxf(v, 0.0f);
}

// xe = relu(edge_x @ We + be); edge_x is [E,2]
__global__ void encode_edges_kernel(const float* __restrict__ ex,
                                    const float* __restrict__ We,
                                    const float* __restrict__ be,
                                    float* __restrict__ xe, int n) {
    int i = blockIdx.x * blockDim.x + threadIdx.x;
    if (i >= n * H) return;
    int r = i >> 6, h = i & 63;
    float v = ex[r * 2] * We[h] + ex[r * 2 + 1] * We[H + h] + be[h];
    xe[i] = fmaxf(v, 0.0f);
}

// ---------------- scatter (segment-sum) ----------------
// per (edge,feature): agg[dst[e]][h] += xsrc[src[e]][h]
__global__ void scatter_add_kernel(const float* __restrict__ xsrc,
                                   const int* __restrict__ src,
                                   const int* __restrict__ dst,
                                   float* __restrict__ agg, int nEdges) {
    int i = blockIdx.x * blockDim.x + threadIdx.x;
    if (i >= nEdges * H) return;
    int e = i >> 6, h = i & 63;
    int s = src[e], d = dst[e];
    unsafeAtomicAdd(&agg[(size_t)d * H + h], xsrc[(size_t)s * H + h]);
}

// degree count (indices are static -> run once, reused by both layers)
__global__ void count_kernel(const int* __restrict__ dst, float* __restrict__ cnt, int n) {
    int i = blockIdx.x * blockDim.x + threadIdx.x;
    if (i < n) unsafeAtomicAdd(&cnt[dst[i]], 1.0f);
}

// ---------------- SAGE dense: out = relu((agg/cnt) @ Wl + bl + xdst @ Wr) ----------------
// 4 waves / block; wave w owns 16x16 output tile (rows = blockIdx*16, cols = w*16).
// V_WMMA_F32_16X16X4_F32: A 16x4 (2 VGPR), B 4x16 (2 VGPR), C/D 16x16 f32 (8 VGPR).
__global__ void sage_wmma_kernel(const float* __restrict__ agg,
                                 const float* __restrict__ cnt,
                                 const float* __restrict__ xdst,
                                 const float* __restrict__ Wl,
                                 const float* __restrict__ bl,
                                 const float* __restrict__ Wr,
                                 float* __restrict__ out, int M) {
    int lane = threadIdx.x & 31;
    int w    = threadIdx.x >> 5;
    int half = lane >> 4;          // 0: K pair {k,k+1}; 1: K pair {k+2,k+3}
    int l15  = lane & 15;
    int row0 = blockIdx.x * 16;
    int col0 = w * 16;
    int arow = row0 + l15;
    if (arow > M - 1) arow = M - 1;            // clamp loads; keep EXEC all-1s for WMMA
    float inv = 1.0f / fmaxf(cnt[arow], 1.0f); // segment mean factor

    const float* aggRow = agg  + (size_t)arow * H;
    const float* dstRow = xdst + (size_t)arow * H;

    v8f acc = {};
#pragma unroll
    for (int k = 0; k < H; k += 4) {
        int ka = k + half * 2;
        // mean(agg) @ Wl
        v2f a, b;
        a.x = aggRow[ka]     * inv;
        a.y = aggRow[ka + 1] * inv;
        b.x = Wl[(size_t)ka * H + col0 + l15];
        b.y = Wl[(size_t)(ka + 1) * H + col0 + l15];
        acc = __builtin_amdgcn_wmma_f32_16x16x4_f32(false, a, false, b, (short)0, acc, false, false);
        // xdst @ Wr
        v2f a2, b2;
        a2.x = dstRow[ka];
        a2.y = dstRow[ka + 1];
        b2.x = Wr[(size_t)ka * H + col0 + l15];
        b2.y = Wr[(size_t)(ka + 1) * H + col0 + l15];
        acc = __builtin_amdgcn_wmma_f32_16x16x4_f32(false, a2, false, b2, (short)0, acc, false, false);
    }

    float bias = bl[col0 + l15];
#pragma unroll
    for (int v = 0; v < 8; v++) {
        int row = row0 + v + half * 8;
        if (row < M) {
            float r = fmaxf(acc[v] + bias, 0.0f);
            out[(size_t)row * H + col0 + l15] = r;
        }
    }
}

// ---------------- graph pooling (segment sums to NUM_GRAPHS) ----------------
__global__ void pool_scatter_kernel(const float* __restrict__ x,
                                    const int* __restrict__ batch,
                                    float* __restrict__ pool,
                                    float* __restrict__ cnt, int n) {
    int i = blockIdx.x * blockDim.x + threadIdx.x;
    if (i >= n * H) return;
    int r = i >> 6, h = i & 63;
    int g = batch[r];
    unsafeAtomicAdd(&pool[g * H + h], x[i]);
    if (h == 0) unsafeAtomicAdd(&cnt[g], 1.0f);
}

// ---------------- head: out = concat(npool/ncnt, epool/ecnt) @ Wout + bout ----------------
// M=64, K=128, N=32 -> 8 tiles, 8 waves, 1 block. WMMA f32 16x16x4, 32 K-steps.
__global__ void head_wmma_kernel(const float* __restrict__ npool,
                                 const float* __restrict__ ncnt,
                                 const float* __restrict__ epool,
                                 const float* __restrict__ ecnt,
                                 const float* __restrict__ Wout,
                                 const float* __restrict__ bout,
                                 float* __restrict__ out) {
    int lane = threadIdx.x & 31;
    int w    = threadIdx.x >> 5;           // 0..7
    int half = lane >> 4;
    int l15  = lane & 15;
    int row0 = (w >> 1) * 16;              // 0,16,32,48
    int col0 = (w & 1) * 16;               // 0,16
    int arow = row0 + l15;                 // M=64 exact
    float ninv = 1.0f / fmaxf(ncnt[arow], 1.0f);
    float einv = 1.0f / fmaxf(ecnt[arow], 1.0f);

    v8f acc = {};
#pragma unroll
    for (int k = 0; k < 2 * H; k += 4) {
        int ka = k + half * 2;             // ka always even -> never straddles the concat seam
        v2f a, b;
        a.x = (ka < H) ? npool[arow * H + ka] * ninv
                       : epool[arow * H + (ka - H)] * einv;
        a.y = (ka + 1 < H) ? npool[arow * H + ka + 1] * ninv
                           : epool[arow * H + (ka + 1 - H)] * einv;
        b.x = Wout[ka * 32 + col0 + l15];
        b.y = Wout[(ka + 1) * 32 + col0 + l15];
        acc = __builtin_amdgcn_wmma_f32_16x16x4_f32(false, a, false, b, (short)0, acc, false, false);
    }
    float bias = bout[col0 + l15];
#pragma unroll
    for (int v = 0; v < 8; v++) {
        int row = row0 + v + half * 8;
        out[row * 32 + col0 + l15] = acc[v] + bias;   // no relu on head
    }
}

// ---------------- launch ----------------
extern "C" void kernel_launch(void* const* d_in, const int* in_sizes, int n_in,
                              void* d_out, int out_size, void* d_ws, size_t ws_size,
                              hipStream_t stream) {
    const float* node_x   = (const float*)d_in[0];
    const float* edge_x   = (const float*)d_in[1];
    const int*   n2e_src  = (const int*)d_in[2];
    const int*   n2e_dst  = (const int*)d_in[3];
    const int*   e2n_src  = (const int*)d_in[4];
    const int*   e2n_dst  = (const int*)d_in[5];
    const int*   node_bat = (const int*)d_in[6];
    const int*   edge_bat = (const int*)d_in[7];
    const float* Wn = (const float*)d_in[8];
    const float* bn = (const float*)d_in[9];
    const float* We = (const float*)d_in[10];
    const float* be = (const float*)d_in[11];
    const float* W1l_n2e = (const float*)d_in[12];
    const float* b1l_n2e = (const float*)d_in[13];
    const float* W1r_n2e = (const float*)d_in[14];
    const float* W1l_e2n = (const float*)d_in[15];
    const float* b1l_e2n = (const float*)d_in[16];
    const float* W1r_e2n = (const float*)d_in[17];
    const float* W2l_n2e = (const float*)d_in[18];
    const float* b2l_n2e = (const float*)d_in[19];
    const float* W2r_n2e = (const float*)d_in[20];
    const float* W2l_e2n = (const float*)d_in[21];
    const float* b2l_e2n = (const float*)d_in[22];
    const float* W2r_e2n = (const float*)d_in[23];
    const float* Wout    = (const float*)d_in[24];
    const float* bout    = (const float*)d_in[25];
    float* out = (float*)d_out;

    // workspace layout (floats)
    float* ws = (float*)d_ws;
    size_t o = 0;
    float* bufA_n = ws + o; o += (size_t)N_NODES * H;   // xn then xn2
    float* bufB_n = ws + o; o += (size_t)N_NODES * H;   // xn1
    float* agg_n  = ws + o; o += (size_t)N_NODES * H;
    float* bufA_e = ws + o; o += (size_t)N_HEDGES * H;  // xe then xe2
    float* bufB_e = ws + o; o += (size_t)N_HEDGES * H;  // xe1
    float* agg_e  = ws + o; o += (size_t)N_HEDGES * H;
    float* cnt_n  = ws + o; o += N_NODES;
    float* cnt_e  = ws + o; o += N_HEDGES;
    float* npool  = ws + o; o += NUM_GRAPHS * H;
    float* ncnt   = ws + o; o += NUM_GRAPHS;
    float* epool  = ws + o; o += NUM_GRAPHS * H;
    float* ecnt   = ws + o; o += NUM_GRAPHS;
    int poolFloats = NUM_GRAPHS * H * 2 + NUM_GRAPHS * 2; // npool..ecnt contiguous

    const int B = 256;
    auto blocks = [](long long n, int b) { return (int)((n + b - 1) / b); };

    // encoders
    encode_nodes_kernel<<<blocks((long long)N_NODES * H, B), B, 0, stream>>>(node_x, Wn, bn, bufA_n, N_NODES);
    encode_edges_kernel<<<blocks((long long)N_HEDGES * H, B), B, 0, stream>>>(edge_x, We, be, bufA_e, N_HEDGES);

    // degree counts (static across both layers)
    fill_zero_kernel<<<blocks(N_NODES, B), B, 0, stream>>>(cnt_n, N_NODES);
    fill_zero_kernel<<<blocks(N_HEDGES, B), B, 0, stream>>>(cnt_e, N_HEDGES);
    count_kernel<<<blocks(N_EDGES, B), B, 0, stream>>>(n2e_dst, cnt_e, N_EDGES);
    count_kernel<<<blocks(N_EDGES, B), B, 0, stream>>>(e2n_dst, cnt_n, N_EDGES);

    const int edgeWork = blocks((long long)N_EDGES * H, B);
    const int tilesE = (N_HEDGES + 15) / 16;
    const int tilesN = (N_NODES + 15) / 16;

    // ---- layer 1: n2e (nodes -> hyperedges), out xe1 = bufB_e
    fill_zero_kernel<<<blocks((long long)N_HEDGES * H, B), B, 0, stream>>>(agg_e, N_HEDGES * H);
    scatter_add_kernel<<<edgeWork, B, 0, stream>>>(bufA_n, n2e_src, n2e_dst, agg_e, N_EDGES);
    sage_wmma_kernel<<<tilesE, 128, 0, stream>>>(agg_e, cnt_e, bufA_e, W1l_n2e, b1l_n2e, W1r_n2e, bufB_e, N_HEDGES);

    // ---- layer 1: e2n (hyperedges -> nodes), out xn1 = bufB_n
    fill_zero_kernel<<<blocks((long long)N_NODES * H, B), B, 0, stream>>>(agg_n, N_NODES * H);
    scatter_add_kernel<<<edgeWork, B, 0, stream>>>(bufA_e, e2n_src, e2n_dst, agg_n, N_EDGES);
    sage_wmma_kernel<<<tilesN, 128, 0, stream>>>(agg_n, cnt_n, bufA_n, W1l_e2n, b1l_e2n, W1r_e2n, bufB_n, N_NODES);

    // ---- layer 2: n2e, out xe2 = bufA_e (xe dead)
    fill_zero_kernel<<<blocks((long long)N_HEDGES * H, B), B, 0, stream>>>(agg_e, N_HEDGES * H);
    scatter_add_kernel<<<edgeWork, B, 0, stream>>>(bufB_n, n2e_src, n2e_dst, agg_e, N_EDGES);
    sage_wmma_kernel<<<tilesE, 128, 0, stream>>>(agg_e, cnt_e, bufB_e, W2l_n2e, b2l_n2e, W2r_n2e, bufA_e, N_HEDGES);

    // ---- layer 2: e2n, out xn2 = bufA_n (xn dead)
    fill_zero_kernel<<<blocks((long long)N_NODES * H, B), B, 0, stream>>>(agg_n, N_NODES * H);
    scatter_add_kernel<<<edgeWork, B, 0, stream>>>(bufB_e, e2n_src, e2n_dst, agg_n, N_EDGES);
    sage_wmma_kernel<<<tilesN, 128, 0, stream>>>(agg_n, cnt_n, bufB_n, W2l_e2n, b2l_e2n, W2r_e2n, bufA_n, N_NODES);

    // ---- pooling
    fill_zero_kernel<<<blocks(poolFloats, B), B, 0, stream>>>(npool, poolFloats);
    pool_scatter_kernel<<<blocks((long long)N_NODES * H, B), B, 0, stream>>>(bufA_n, node_bat, npool, ncnt, N_NODES);
    pool_scatter_kernel<<<blocks((long long)N_HEDGES * H, B), B, 0, stream>>>(bufA_e, edge_bat, epool, ecnt, N_HEDGES);

    // ---- head
    head_wmma_kernel<<<1, 256, 0, stream>>>(npool, ncnt, epool, ecnt, Wout, bout, out);
}